// FineMatching_9405978378734
// MI455X (gfx1250) — compile-verified
//
#include <hip/hip_runtime.h>
#include <hip/hip_bf16.h>

typedef float v2f __attribute__((ext_vector_type(2)));
typedef float v8f __attribute__((ext_vector_type(8)));
typedef int   v4i __attribute__((ext_vector_type(4)));

#define NC 128          // channels
#define NP 64           // ws*ws
#define NTHREADS 512    // 16 wave32 waves

#if defined(__has_builtin)
#if __has_builtin(__builtin_amdgcn_global_load_async_to_lds_b128)
#define HAVE_ASYNC_LDS 1
#endif
#endif

// Wm[o][y]: 1D bilinear sampling matrix, p = (o+0.5)*7/8, i0=floor(p), w=p-i0
__device__ __forceinline__ float wm_entry(int o, int y) {
  float p = ((float)o + 0.5f) * 0.875f;
  float fi = floorf(p);
  int i0 = (int)fi;
  float w = p - fi;
  float r = 0.0f;
  if (y == i0) r += 1.0f - w;
  if (y == i0 + 1) r += w;
  return r;
}

__global__ __launch_bounds__(NTHREADS) void fine_matching_kernel(
    const float* __restrict__ x0, const float* __restrict__ x1,
    float* __restrict__ out, int M) {
  // lA: X0 during stage1; reused for T [0..4096) and sim [4096..8192) later
  __shared__ __align__(16) float lA[NC * NP];
  __shared__ __align__(16) float lB[NC * NP];      // X1
  __shared__ __align__(16) float lK[NP * NP];      // K = kron(Wm, Wm), row-major
  __shared__ __align__(16) float lG[NP * NP];      // G = X0^T X1
  __shared__ float rmax_[NP], rsum_[NP], cmax_[NP], csum_[NP];
  __shared__ float bestv[16];
  __shared__ int   besti[16];

  const int m    = blockIdx.x;
  const int t    = threadIdx.x;
  const int w    = t >> 5;
  const int lane = t & 31;
  const int li   = lane & 15;      // row/col index within 16x16 tile
  const int half = lane >> 4;      // 0 or 1 (K pairing for f32 fragments)
  const int it   = (w >> 2) << 4;  // tile row base
  const int jt   = (w & 3)  << 4;  // tile col base

  // ---- Stage 0: stream X0/X1 into LDS, build K64 (overlapped) ----
  {
    const float4* g0 = (const float4*)(x0 + (size_t)m * (NC * NP));
    const float4* g1 = (const float4*)(x1 + (size_t)m * (NC * NP));
    float4* s0 = (float4*)lA;
    float4* s1 = (float4*)lB;
#ifdef HAVE_ASYNC_LDS
    typedef __attribute__((address_space(1))) v4i* gas1_t;
    typedef __attribute__((address_space(3))) v4i* las3_t;
#pragma unroll
    for (int s = 0; s < 4; ++s) {
      float4* p0 = (float4*)(g0 + t + s * NTHREADS);   // drop const (generic)
      float4* p1 = (float4*)(g1 + t + s * NTHREADS);
      __builtin_amdgcn_global_load_async_to_lds_b128(
          (gas1_t)p0, (las3_t)(s0 + t + s * NTHREADS), 0, 0);
      __builtin_amdgcn_global_load_async_to_lds_b128(
          (gas1_t)p1, (las3_t)(s1 + t + s * NTHREADS), 0, 0);
    }
#else
#pragma unroll
    for (int s = 0; s < 4; ++s) {
      s0[t + s * NTHREADS] = g0[t + s * NTHREADS];
      s1[t + s * NTHREADS] = g1[t + s * NTHREADS];
    }
#endif
    // K64 VALU work overlaps the in-flight async copies
#pragma unroll
    for (int s = 0; s < 8; ++s) {
      int e = t * 8 + s;                 // 0..4095
      int i = e >> 6, j = e & 63;        // K[i][j] = Wm[i/8][j/8]*Wm[i%8][j%8]
      lK[e] = wm_entry(i >> 3, j >> 3) * wm_entry(i & 7, j & 7);
    }
#ifdef HAVE_ASYNC_LDS
    asm volatile("s_wait_asynccnt 0" ::: "memory");
#endif
  }
  __syncthreads();

  // ---- Stage 1: G = X0^T X1  (contract over C=128), f32 WMMA 16x16x4 ----
  v8f acc = {};
#pragma unroll
  for (int k = 0; k < NC; k += 4) {
    int c0 = k + 2 * half;
    v2f a, b;
    a.x = lA[c0 * NP + it + li];        // A[i, c] = X0[c][it+i]
    a.y = lA[(c0 + 1) * NP + it + li];
    b.x = lB[c0 * NP + jt + li];        // B[c, j] = X1[c][jt+j]
    b.y = lB[(c0 + 1) * NP + jt + li];
    acc = __builtin_amdgcn_wmma_f32_16x16x4_f32(false, a, false, b,
                                                (short)0, acc, false, false);
  }
#pragma unroll
  for (int r = 0; r < 8; ++r)
    lG[(it + 8 * half + r) * NP + jt + li] = acc[r];
  __syncthreads();

  // ---- Stage 2: T = K * G  (contract over 64) ; T overwrites X0 region ----
  float* lT = lA;
  v8f acc2 = {};
#pragma unroll
  for (int k = 0; k < NP; k += 4) {
    int i0 = k + 2 * half;
    v2f a = *(const v2f*)&lK[(it + li) * NP + i0];  // A[o, i] = K[it+o][i]
    v2f b;
    b.x = lG[i0 * NP + jt + li];                    // B[i, j] = G[i][jt+j]
    b.y = lG[(i0 + 1) * NP + jt + li];
    acc2 = __builtin_amdgcn_wmma_f32_16x16x4_f32(false, a, false, b,
                                                 (short)0, acc2, false, false);
  }
#pragma unroll
  for (int r = 0; r < 8; ++r)
    lT[(it + 8 * half + r) * NP + jt + li] = acc2[r];
  __syncthreads();

  // ---- Stage 3: sim = (T * K^T) / C ----
  float* lS = lA + NP * NP;   // sim lives in second half of lA
  v8f acc3 = {};
#pragma unroll
  for (int k = 0; k < NP; k += 4) {
    int j0 = k + 2 * half;
    v2f a = *(const v2f*)&lT[(it + li) * NP + j0];  // A[o, j] = T[it+o][j]
    v2f b = *(const v2f*)&lK[(jt + li) * NP + j0];  // B[j, p] = K[jt+p][j]
    acc3 = __builtin_amdgcn_wmma_f32_16x16x4_f32(false, a, false, b,
                                                 (short)0, acc3, false, false);
  }
  const float invC = 1.0f / 128.0f;
#pragma unroll
  for (int r = 0; r < 8; ++r)
    lS[(it + 8 * half + r) * NP + jt + li] = acc3[r] * invC;
  __syncthreads();

  // ---- Stage 4: row (axis=-1) and column (axis=-2) softmax statistics ----
  {
    int r = t >> 3;              // row (or column) index 0..63
    int s8 = (t & 7) * 8;        // 8-element segment
    // rows
    float mx = -3.402823466e38f;
    for (int q = 0; q < 8; ++q) mx = fmaxf(mx, lS[r * NP + s8 + q]);
    for (int d = 1; d <= 4; d <<= 1) mx = fmaxf(mx, __shfl_xor(mx, d, 32));
    float se = 0.0f;
    for (int q = 0; q < 8; ++q) se += __expf(lS[r * NP + s8 + q] - mx);
    for (int d = 1; d <= 4; d <<= 1) se += __shfl_xor(se, d, 32);
    if ((t & 7) == 0) { rmax_[r] = mx; rsum_[r] = se; }
    // columns
    float cx = -3.402823466e38f;
    for (int q = 0; q < 8; ++q) cx = fmaxf(cx, lS[(s8 + q) * NP + r]);
    for (int d = 1; d <= 4; d <<= 1) cx = fmaxf(cx, __shfl_xor(cx, d, 32));
    float cs = 0.0f;
    for (int q = 0; q < 8; ++q) cs += __expf(lS[(s8 + q) * NP + r] - cx);
    for (int d = 1; d <= 4; d <<= 1) cs += __shfl_xor(cs, d, 32);
    if ((t & 7) == 0) { cmax_[r] = cx; csum_[r] = cs; }
  }
  __syncthreads();

  // ---- Stage 5: heatmap = softmax_col * softmax_row ; write + argmax ----
  float bv = -3.402823466e38f;
  int   bi = 0;
  {
    int o  = t >> 3;
    int pb = (t & 7) * 8;
    float ro = rmax_[o], rs = rsum_[o];
    float hq[8];
#pragma unroll
    for (int q = 0; q < 8; ++q) {
      int p = pb + q;
      float s = lS[o * NP + p];
      float h = __expf(2.0f * s - ro - cmax_[p]) / (rs * csum_[p]);
      hq[q] = h;
      int e = o * NP + p;
      if (h > bv) { bv = h; bi = e; }   // strict > keeps first index
    }
    float4* ho = (float4*)(out + (size_t)m * (NP * NP) + o * NP + pb);
    ho[0] = make_float4(hq[0], hq[1], hq[2], hq[3]);
    ho[1] = make_float4(hq[4], hq[5], hq[6], hq[7]);
  }
  for (int d = 1; d < 32; d <<= 1) {
    float ov = __shfl_xor(bv, d, 32);
    int   oi = __shfl_xor(bi, d, 32);
    if (ov > bv || (ov == bv && oi < bi)) { bv = ov; bi = oi; }
  }
  if (lane == 0) { bestv[w] = bv; besti[w] = bi; }
  __syncthreads();

  if (t == 0) {
    float fbv = bestv[0]; int fbi = besti[0];
    for (int u = 1; u < 16; ++u) {
      if (bestv[u] > fbv || (bestv[u] == fbv && besti[u] < fbi)) {
        fbv = bestv[u]; fbi = besti[u];
      }
    }
    int ii = fbi >> 6, jj = fbi & 63;
    float* idxo = out + (size_t)M * (NP * NP);   // indices [3, M] (as float)
    idxo[m]         = (float)m;
    idxo[M + m]     = (float)ii;
    idxo[2 * M + m] = (float)jj;
    float* b0 = idxo + 3 * (size_t)M;            // biases0 [M, 2]
    b0[2 * m + 0] = (float)(ii & 7) - 3.5f;
    b0[2 * m + 1] = (float)(ii >> 3) - 3.5f;
    float* b1 = b0 + 2 * (size_t)M;              // biases1 [M, 2]
    b1[2 * m + 0] = (float)(jj & 7) - 3.5f;
    b1[2 * m + 1] = (float)(jj >> 3) - 3.5f;
  }
}

extern "C" void kernel_launch(void* const* d_in, const int* in_sizes, int n_in,
                              void* d_out, int out_size, void* d_ws, size_t ws_size,
                              hipStream_t stream) {
  const float* x0 = (const float*)d_in[0];
  const float* x1 = (const float*)d_in[1];
  float* out = (float*)d_out;
  int M = in_sizes[0] / (NC * NP);   // 5000
  fine_matching_kernel<<<M, NTHREADS, 0, stream>>>(x0, x1, out, M);
}